// MultiHeadAttentionQuantum_65481071404431
// MI455X (gfx1250) — compile-verified
//
#include <hip/hip_runtime.h>
#include <hip/hip_bf16.h>

// y[m, o] = sum_e cos(x[m, e] + theta[e]) * w[o, e]
//   M = BATCH*SEQ = 1,048,576 tokens, E = 16.
// Memory-bound: 128 MiB traffic -> ~5.8 us @ 23.3 TB/s. f32 WMMA keeps us on
// the bandwidth roof at full precision (no benefit to f16/fp8: traffic is f32).
//
// One wave32 per 16-token tile: Y(16x16) = Z(16x16) @ W^T(16x16), computed as
// four chained V_WMMA_F32_16X16X4_F32 over K-chunks of 4.

typedef __attribute__((ext_vector_type(2))) float v2f;
typedef __attribute__((ext_vector_type(8))) float v8f;

__global__ __launch_bounds__(256) void qattn_cos_wmma_kernel(
    const float* __restrict__ x,      // [M, 16]
    const float* __restrict__ theta,  // [16]
    const float* __restrict__ w,      // [16, 16] row-major, w[o][e]
    float* __restrict__ out,          // [M, 16]
    int num_tiles)
{
  const int lane = (int)(threadIdx.x & 31u);   // wave32
  const int wave = (int)(threadIdx.x >> 5);    // 8 waves / block
  const int tile = (int)blockIdx.x * 8 + wave; // uniform across the wave
  if (tile >= num_tiles) return;               // uniform branch; EXEC stays all-1s

  const int idx  = lane & 15;   // A: row M; B/store: col N
  const int half = lane >> 4;   // 0 = lanes 0-15, 1 = lanes 16-31

  const long tile_base = (long)tile * 16;

  // ---- B = W^T chunks. ISA f32 B layout (4x16, KxN): for chunk c,
  // lane (n + 16h) vgpr0/vgpr1 = B[4c+2h][n], B[4c+2h+1][n] = w[n][4c+2h +{0,1}].
  v2f B[4];
  {
    const float* wrow = w + idx * 16;
    #pragma unroll
    for (int c = 0; c < 4; ++c)
      B[c] = *(const v2f*)(wrow + 4 * c + 2 * half);   // 8B-aligned pair
  }

  // ---- theta pairs matching the A layout (uniform per half-wave)
  v2f Th[4];
  #pragma unroll
  for (int c = 0; c < 4; ++c)
    Th[c] = *(const v2f*)(theta + 4 * c + 2 * half);

  // ---- A = Z chunks. ISA f32 A layout (16x4, MxK): for chunk c,
  // lane (m + 16h) vgpr0/vgpr1 = A[m][4c+2h], A[m][4c+2h+1].
  const float* xrow = x + (tile_base + idx) * 16;
  v2f A[4];
  #pragma unroll
  for (int c = 0; c < 4; ++c)
    A[c] = *(const v2f*)(xrow + 4 * c + 2 * half);     // global_load_b64

  // z = cos(x + theta); v_cos_f32 takes revolutions -> pre-scale by 1/(2*pi)
  const float r2pi = 0.15915494309189535f;
  #pragma unroll
  for (int c = 0; c < 4; ++c) {
    A[c].x = __builtin_amdgcn_cosf((A[c].x + Th[c].x) * r2pi);
    A[c].y = __builtin_amdgcn_cosf((A[c].y + Th[c].y) * r2pi);
  }

  // ---- accumulate 4x V_WMMA_F32_16X16X4_F32
  v8f acc = {};
  #pragma unroll
  for (int c = 0; c < 4; ++c) {
    acc = __builtin_amdgcn_wmma_f32_16x16x4_f32(
        /*neg_a=*/false, A[c], /*neg_b=*/false, B[c],
        /*c_mod=*/(short)0, acc, /*reuse_a=*/false, /*reuse_b=*/false);
  }

  // ---- store: acc[j] on lane (n + 16h) is Y[j + 8h][n]
  // -> per j, lanes 0-15 write one coalesced 64B row, lanes 16-31 another.
  float* orow = out + (tile_base + half * 8) * 16 + idx;
  #pragma unroll
  for (int j = 0; j < 8; ++j)
    orow[j * 16] = acc[j];
}

extern "C" void kernel_launch(void* const* d_in, const int* in_sizes, int n_in,
                              void* d_out, int out_size, void* d_ws, size_t ws_size,
                              hipStream_t stream) {
  const float* x     = (const float*)d_in[0];  // [B, S, 16] f32
  const float* theta = (const float*)d_in[1];  // [16] f32
  const float* w_out = (const float*)d_in[2];  // [16, 16] f32
  float* out = (float*)d_out;                  // [B, S, 16] f32

  const long m_total   = (long)in_sizes[0] / 16;   // tokens
  const int  num_tiles = (int)(m_total / 16);      // 65,536 for B=128,S=8192
  const int  blocks    = (num_tiles + 7) / 8;      // 8 waves per 256-thread block

  qattn_cos_wmma_kernel<<<blocks, 256, 0, stream>>>(x, theta, w_out, out, num_tiles);
}